// PatchShifting_82721070121355
// MI455X (gfx1250) — compile-verified
//
#include <hip/hip_runtime.h>
#include <math.h>

// Problem constants (fixed by the reference: x = [16, 3, 512, 1024] fp32)
#define B_   16
#define C_   3
#define H_   512
#define W_   1024
#define HW_  (H_ * W_)          // 524288
#define NPLANES_ (B_ * C_)      // 48
#define OUTC_ 15

// TDM staging: 32KB chunks, 8-byte elements
#define CHUNK_BYTES 32768
#define CHUNK_ELEMS (CHUNK_BYTES / 8)            // 4096 (fits tile_dim0 u16, mult of 4B)
#define PLANE_BYTES (HW_ * 4)                    // 2 MB per plane
#define CHUNKS_PER_PLANE (PLANE_BYTES / CHUNK_BYTES)   // 64
#define TOTAL_CHUNKS (NPLANES_ * CHUNKS_PER_PLANE)     // 3072

typedef unsigned int u32x4 __attribute__((ext_vector_type(4)));
typedef int          i32x4 __attribute__((ext_vector_type(4)));
typedef int          i32x8 __attribute__((ext_vector_type(8)));

// ---------------------------------------------------------------------------
// Kernel 1: replicate x into output channel blocks {0..2, 6..8, 9..11, 12..14}
// via the CDNA5 Tensor Data Mover. One wave per 32KB chunk:
//   TENSOR_LOAD_TO_LDS (RT -> warms the 192MB L2 with x)
//   s_wait_tensorcnt 0
//   4x TENSOR_STORE_FROM_LDS (cpol TH=NT -> stream write-once output past L2)
// D# built per CDNA5 ISA:
//   group0: [1:0]=count=1 | lds_addr | global_addr[56:0] | type=2 (bits 127:126)
//   group1: data_size=3 (8B), tensor_dim0=4096, tensor_dim1=1,
//           tile_dim0=4096, tile_dim1=1, stride0=4096; groups 2/3 zero (<=2D)
// amdgpu-toolchain builtin form (6 args): (u32x4, i32x8, i32x4, i32x4, i32x8, cpol)
// ---------------------------------------------------------------------------
__global__ __launch_bounds__(32) void replicate_tdm(const float* __restrict__ x,
                                                    float* __restrict__ out) {
    __shared__ char buf[CHUNK_BYTES];
    // Flat LDS addresses: low 32 bits of the generic pointer = LDS byte offset.
    unsigned lds_off = (unsigned)(size_t)(void*)buf;

    unsigned chunk = blockIdx.x;                  // 0 .. TOTAL_CHUNKS-1 (uniform)
    unsigned p = chunk / CHUNKS_PER_PLANE;        // source plane 0..47
    unsigned r = chunk - p * CHUNKS_PER_PLANE;    // chunk within plane
    unsigned b = p / 3u;
    unsigned c = p - 3u * b;

    unsigned long long src = (unsigned long long)(size_t)x
                           + (unsigned long long)chunk * CHUNK_BYTES;

    // D# group 1 (shared by load and all stores)
    i32x8 g1;
    g1[0] = (int)(3u << 16);                      // wg_mask=0 | data_size=3 (8B)
    g1[1] = (int)((unsigned)CHUNK_ELEMS << 16);   // atomic_barrier_addr=0 | tensor_dim0[15:0]
    g1[2] = (int)(1u << 16);                      // tensor_dim0[31:16]=0 | tensor_dim1[15:0]=1
    g1[3] = (int)((unsigned)CHUNK_ELEMS << 16);   // tensor_dim1[31:16]=0 | tile_dim0=4096
    g1[4] = 1;                                    // tile_dim1=1 | tile_dim2=0
    g1[5] = (int)CHUNK_ELEMS;                     // tensor_dim0_stride[31:0]
    g1[6] = 0;                                    // stride0[47:32]=0 | stride1[15:0]=0
    g1[7] = 0;                                    // stride1[47:16]=0
    i32x4 z4 = {0, 0, 0, 0};
    i32x8 z8 = {0, 0, 0, 0, 0, 0, 0, 0};

    // Load descriptor: global x chunk -> LDS (RT policy: keep x in L2)
    u32x4 g0l;
    g0l[0] = 1u;                                  // count=1, user mode
    g0l[1] = lds_off;
    g0l[2] = (unsigned)(src & 0xFFFFFFFFull);
    g0l[3] = (unsigned)((src >> 32) & 0x01FFFFFFull) | (2u << 30);  // type=2
    __builtin_amdgcn_tensor_load_to_lds(g0l, g1, z4, z4, z8, 0);
    __builtin_amdgcn_s_wait_tensorcnt(0);         // LDS tile resident

    // 4 stores into the identity channel blocks (out channels {0,6,9,12}+c)
    const unsigned K[4] = {0u, 6u, 9u, 12u};
#pragma unroll
    for (int k = 0; k < 4; ++k) {
        unsigned plane_out = b * OUTC_ + K[k] + c;
        unsigned long long dst = (unsigned long long)(size_t)out
                               + (unsigned long long)plane_out * PLANE_BYTES
                               + (unsigned long long)r * CHUNK_BYTES;
        u32x4 g0s;
        g0s[0] = 1u;
        g0s[1] = lds_off;
        g0s[2] = (unsigned)(dst & 0xFFFFFFFFull);
        g0s[3] = (unsigned)((dst >> 32) & 0x01FFFFFFull) | (2u << 30);
        __builtin_amdgcn_tensor_store_from_lds(g0s, g1, z4, z4, z8, 1 /* TH=NT */);
    }
    __builtin_amdgcn_s_wait_tensorcnt(0);         // drain before wave exit
}

// ---------------------------------------------------------------------------
// Kernel 2: spherical y-rotation (beta = 15 deg) via inverse warp + bilinear.
// One thread per (h,w): trig computed ONCE, reused for all 48 (b,c) planes.
// Gather taps are RT loads (hit the L2-resident input); output stores are NT.
// global_prefetch_b8 stages the next plane's gather line.
// ---------------------------------------------------------------------------
__global__ __launch_bounds__(256) void rotate_sphere_y(const float* __restrict__ x,
                                                       float* __restrict__ out) {
    int hw = blockIdx.x * 256 + threadIdx.x;      // over H*W
    if (hw >= HW_) return;
    int h = hw >> 10;                             // W_ = 1024
    int w = hw & (W_ - 1);

    const float PI = 3.14159265358979323846f;
    float th = ((float)h + 0.5f) * (PI / (float)H_);
    float ph = ((float)w + 0.5f) * (2.0f * PI / (float)W_) - PI;

    float st, ct, sp, cp;
    sincosf(th, &st, &ct);
    sincosf(ph, &sp, &cp);

    float vx = st * cp;
    float vy = st * sp;
    float vz = ct;

    // Ry(-15deg)
    const float cb = 0.96592582628906828675f;
    const float sb = 0.25881904510252076235f;
    float ux = cb * vx - sb * vz;
    float uy = vy;
    float uz = sb * vx + cb * vz;
    uz = fminf(1.0f, fmaxf(-1.0f, uz));

    float th_s = acosf(uz);
    float ph_s = atan2f(uy, ux);

    float fi = th_s * ((float)H_ / PI) - 0.5f;
    float fj = (ph_s + PI) * ((float)W_ / (2.0f * PI)) - 0.5f;

    float i0f = floorf(fi);
    float j0f = floorf(fj);
    float wi = fi - i0f;                          // weights BEFORE clamp (matches ref)
    float wj = fj - j0f;

    int i0 = (int)i0f;
    i0 = min(max(i0, 0), H_ - 1);
    int i1 = min(i0 + 1, H_ - 1);                 // clamp latitude
    int j0 = ((int)j0f) & (W_ - 1);               // wrap longitude (pow2 AND == mod)
    int j1 = (j0 + 1) & (W_ - 1);

    float w00 = (1.0f - wi) * (1.0f - wj);
    float w01 = (1.0f - wi) * wj;
    float w10 = wi * (1.0f - wj);
    float w11 = wi * wj;

    int o00 = i0 * W_ + j0;
    int o01 = i0 * W_ + j1;
    int o10 = i1 * W_ + j0;
    int o11 = i1 * W_ + j1;

#pragma unroll 4
    for (int p = 0; p < NPLANES_; ++p) {
        const float* __restrict__ xp = x + (size_t)p * HW_;
        if (p + 1 < NPLANES_) {
            __builtin_prefetch(xp + HW_ + o00, 0, 1);   // global_prefetch_b8
        }
        float v = xp[o00] * w00 + xp[o01] * w01
                + xp[o10] * w10 + xp[o11] * w11;

        int b = p / C_;
        int c = p - b * C_;
        size_t oidx = (size_t)(b * OUTC_ + 3 + c) * HW_ + (size_t)hw;  // ch 3..5
        __builtin_nontemporal_store(v, &out[oidx]);
    }
}

extern "C" void kernel_launch(void* const* d_in, const int* in_sizes, int n_in,
                              void* d_out, int out_size, void* d_ws, size_t ws_size,
                              hipStream_t stream) {
    (void)in_sizes; (void)n_in; (void)out_size; (void)d_ws; (void)ws_size;

    const float* x = (const float*)d_in[0];
    float* out     = (float*)d_out;

    // Pass 1: TDM bulk replication (also warms L2 with x).
    replicate_tdm<<<TOTAL_CHUNKS, 32, 0, stream>>>(x, out);

    // Pass 2: rotated copy into channels 3..5 (gathers hit L2-resident x).
    rotate_sphere_y<<<(HW_ + 255) / 256, 256, 0, stream>>>(x, out);
}